// RecurrentContext_89386859365029
// MI455X (gfx1250) — compile-verified
//
#include <hip/hip_runtime.h>
#include <hip/hip_bf16.h>

typedef __attribute__((ext_vector_type(2))) float v2f;
typedef __attribute__((ext_vector_type(8))) float v8f;

#define Bsz 64
#define Ssz 128
#define Tsz 64
#define Fsz 100

// ---------------------------------------------------------------------------
// Phase 1: xW[b,s,t,f] = sum_k x[b,s,t,k] * W[k,f] + bias[f]
// One block per (b,s). 28 waves, one 16x16 tile each (4 M-tiles x 7 N-tiles).
// fp32 WMMA 16x16x4, K=100 = 25 * 4 exactly (no K guard needed).
// ---------------------------------------------------------------------------
__global__ __launch_bounds__(896) void
xw_wmma_kernel(const float* __restrict__ x, const float* __restrict__ W,
               const float* __restrict__ bias, float* __restrict__ xW)
{
    __shared__ float ldsX[Tsz * 101];   // 64 rows x K=100, stride 101 (conflict-free)

    const int    bs   = blockIdx.x;         // 0 .. B*S-1
    const int    tid  = threadIdx.x;
    const size_t base = (size_t)bs * (Tsz * Fsz);

    // Stage x tile into LDS (coalesced).
    for (int i = tid; i < Tsz * Fsz; i += 896) {
        int t = i / Fsz;
        int k = i - t * Fsz;
        ldsX[t * 101 + k] = x[base + i];
    }
    __syncthreads();

    const int wave  = tid >> 5;
    const int lane  = tid & 31;
    const int mTile = wave / 7;             // 0..3
    const int nTile = wave - mTile * 7;     // 0..6
    const int m0    = mTile * 16;
    const int n0    = nTile * 16;
    const int l16   = lane & 15;
    const int halfK = (lane >> 4) * 2;      // 0 or 2
    const int mRow  = m0 + l16;             // A: M = lane%16
    const int nCol  = n0 + l16;             // B/D: N = lane%16
    const bool nOK  = (nCol < Fsz);
    const int nColC = nOK ? nCol : (Fsz - 1);

    v8f acc = {};
    for (int k0 = 0; k0 < Fsz; k0 += 4) {
        v2f av, bv;
        // A fragment: a[v] = A[M = lane%16, K = k0 + 2*(lane/16) + v]
        av[0] = ldsX[mRow * 101 + k0 + halfK];
        av[1] = ldsX[mRow * 101 + k0 + halfK + 1];
        // B fragment: b[v] = B[K = k0 + 2*(lane/16) + v, N = lane%16]
        float b0 = W[(k0 + halfK) * Fsz + nColC];
        float b1 = W[(k0 + halfK + 1) * Fsz + nColC];
        bv[0] = nOK ? b0 : 0.0f;
        bv[1] = nOK ? b1 : 0.0f;
        acc = __builtin_amdgcn_wmma_f32_16x16x4_f32(false, av, false, bv,
                                                    (short)0, acc, false, false);
    }

    // D layout: lane%16 = N, VGPR v -> M = v + 8*(lane/16)
    if (nOK) {
        float badd = bias[nCol];
#pragma unroll
        for (int v = 0; v < 8; ++v) {
            int t = m0 + v + 8 * (lane >> 4);
            xW[base + (size_t)t * Fsz + nCol] = acc[v] + badd;
        }
    }
}

// ---------------------------------------------------------------------------
// Phase 2: sequential scan. One block per batch element (independent carry h).
// 256 threads, h[100] lives in LDS for all 128 steps.
// ---------------------------------------------------------------------------
__global__ __launch_bounds__(256) void
seq_kernel(const float* __restrict__ x, const float* __restrict__ xW,
           const float* __restrict__ Wc, const float* __restrict__ u,
           const float* __restrict__ Wx, const float* __restrict__ Wh,
           const float* __restrict__ bg, float* __restrict__ out)
{
    __shared__ float h[Fsz];
    __shared__ float cvec[Fsz];
    __shared__ float aw[Tsz];
    __shared__ float att[Fsz];
    __shared__ float zg[Fsz], rg[Fsz], g3x[Fsz], g3h[Fsz];
    __shared__ float sumEps;

    const int b   = blockIdx.x;     // 0..63
    const int tid = threadIdx.x;

    if (tid < Fsz) h[tid] = 0.0f;
    __syncthreads();

    for (int s = 0; s < Ssz; ++s) {
        const size_t base = ((size_t)b * Ssz + s) * (Tsz * Fsz);
        const float* xs  = x  + base;
        const float* xWs = xW + base;

        // c = h @ W_context   (coalesced over f)
        if (tid < Fsz) {
            float acc = 0.0f;
            for (int k = 0; k < Fsz; ++k) acc += h[k] * Wc[k * Fsz + tid];
            cvec[tid] = acc;
        }
        __syncthreads();

        // ait[t] = sum_f tanh(xW[t,f] + c[f]) * u[f]; 4 lanes per t.
        {
            int t  = tid >> 2;
            int l4 = tid & 3;
            float p = 0.0f;
            for (int f = l4; f < Fsz; f += 4) {
                float uv = tanhf(xWs[t * Fsz + f] + cvec[f]);
                p += uv * u[f];
            }
            p += __shfl_xor(p, 1);
            p += __shfl_xor(p, 2);
            if (l4 == 0) aw[t] = __expf(p);
        }
        __syncthreads();

        // sum over T=64 (exact reference form: no max-subtract, +EPS)
        if (tid < 32) {
            float s2 = aw[tid] + aw[tid + 32];
            s2 += __shfl_xor(s2, 16);
            s2 += __shfl_xor(s2, 8);
            s2 += __shfl_xor(s2, 4);
            s2 += __shfl_xor(s2, 2);
            s2 += __shfl_xor(s2, 1);
            if (tid == 0) sumEps = s2 + 1e-7f;
        }
        __syncthreads();
        float invSum = 1.0f / sumEps;
        if (tid < Tsz) aw[tid] *= invSum;
        __syncthreads();

        // attended[f] = sum_t x[t,f] * a[t]   (coalesced over f)
        if (tid < Fsz) {
            float acc = 0.0f;
            for (int t = 0; t < Tsz; ++t) acc += xs[t * Fsz + tid] * aw[t];
            att[tid] = acc;
        }
        __syncthreads();

        // GRU gates: gx = attended@Wx + bg, gh = h@Wh  (300 outputs)
        for (int j = tid; j < 3 * Fsz; j += 256) {
            float gx = bg[j];
            float gh = 0.0f;
            for (int k = 0; k < Fsz; ++k) {
                gx += att[k] * Wx[k * (3 * Fsz) + j];
                gh += h[k]   * Wh[k * (3 * Fsz) + j];
            }
            if (j < Fsz) {
                zg[j] = 1.0f / (1.0f + __expf(-(gx + gh)));
            } else if (j < 2 * Fsz) {
                rg[j - Fsz] = 1.0f / (1.0f + __expf(-(gx + gh)));
            } else {
                g3x[j - 2 * Fsz] = gx;
                g3h[j - 2 * Fsz] = gh;
            }
        }
        __syncthreads();

        // h_new = (1-z)*h + z*tanh(gx3 + r*gh3); emit and carry.
        if (tid < Fsz) {
            float ht = tanhf(g3x[tid] + rg[tid] * g3h[tid]);
            float z  = zg[tid];
            float hn = (1.0f - z) * h[tid] + z * ht;
            out[((size_t)b * Ssz + s) * Fsz + tid] = hn;
            h[tid] = hn;
        }
        __syncthreads();
    }
}

extern "C" void kernel_launch(void* const* d_in, const int* in_sizes, int n_in,
                              void* d_out, int out_size, void* d_ws, size_t ws_size,
                              hipStream_t stream) {
    (void)in_sizes; (void)n_in; (void)out_size; (void)ws_size;

    const float* x    = (const float*)d_in[0];
    const float* W    = (const float*)d_in[1];
    const float* Wc   = (const float*)d_in[2];
    const float* bias = (const float*)d_in[3];
    const float* u    = (const float*)d_in[4];
    const float* Wx   = (const float*)d_in[5];
    const float* Wh   = (const float*)d_in[6];
    const float* bg   = (const float*)d_in[7];
    float* out = (float*)d_out;

    // Workspace: xW = B*S*T*F floats = 209,715,200 bytes.
    float* xW = (float*)d_ws;

    xw_wmma_kernel<<<Bsz * Ssz, 896, 0, stream>>>(x, W, bias, xW);
    seq_kernel<<<Bsz, 256, 0, stream>>>(x, xW, Wc, u, Wx, Wh, bg, out);
}